// MyLSTM_19980187861904
// MI455X (gfx1250) — compile-verified
//
#include <hip/hip_runtime.h>

// ---------------------------------------------------------------------------
// 2-layer LSTM (hidden=4, input=3) + per-step MLP head on CDNA5 using
// V_WMMA_F32_16X16X4_F32 (exact-shape K=4 f32 WMMA).
//
// One wave = 16 batch rows. Per timestep, per layer:
//   D = A_x (16x4) x W_ih'^T (4x16) + bias'   (wmma #1, C = bias splat)
//   D = A_h (16x4) x W_hh'^T (4x16) + D       (wmma #2, C = D)
// The sigmoid-vs-tanh gate split is per-column == per-lane, and the sigmoid
// identity sig(v) = 0.5 + 0.5*tanh(v/2) lets the 0.5 pre-scale be folded
// into the (per-column) weights/bias, so each activation is exactly
// v_tanh_f32 + v_fma. A 16x16 LDS transpose (2addr-merged DS ops) converts
// D layout back to the A layout for the recurrence. t is unrolled 4x:
// three global_load_b128 per lane fetch 4 timesteps of x, four y results
// leave via one global_store_b128.
// ---------------------------------------------------------------------------

typedef float v2f __attribute__((ext_vector_type(2)));
typedef float v4f __attribute__((ext_vector_type(4)));
typedef float v8f __attribute__((ext_vector_type(8)));

#define TSEQ 1024
#define LOG2E 1.4426950408889634f

#if __has_builtin(__builtin_amdgcn_tanhf)
#define HAVE_HW_TANH 1
#else
#define HAVE_HW_TANH 0
#endif

__device__ __forceinline__ float fast_tanh(float v) {
#if HAVE_HW_TANH
    return __builtin_amdgcn_tanhf(v);          // single v_tanh_f32
#else
    float e = __builtin_amdgcn_exp2f(v * (-2.0f * LOG2E));
    return __fmaf_rn(2.0f, __builtin_amdgcn_rcpf(1.0f + e), -1.0f);
#endif
}

__global__ void __launch_bounds__(256)
lstm2_wmma_kernel(const float* __restrict__ x,
                  const float* __restrict__ W_ih0, const float* __restrict__ W_hh0,
                  const float* __restrict__ b_ih0, const float* __restrict__ b_hh0,
                  const float* __restrict__ W_ih1, const float* __restrict__ W_hh1,
                  const float* __restrict__ b_ih1, const float* __restrict__ b_hh1,
                  const float* __restrict__ W1, const float* __restrict__ b1,
                  const float* __restrict__ W2, const float* __restrict__ b2,
                  float* __restrict__ out)
{
    // per-wave 16x16 transpose scratch (rows of 16 -> 8B-aligned pair loads)
    __shared__ float xposeS[8][16 * 16];

    const int lane = threadIdx.x & 31;
    const int wv   = threadIdx.x >> 5;
    const int n    = lane & 15;   // B/C/D column (gate); A row (batch); gather row
    const int half = lane >> 4;   // K pair {0,1} vs {2,3}
    const int k0   = 2 * half;

    const int batch = blockIdx.x * 128 + wv * 16 + n;
    const float* __restrict__ xb = x + (size_t)batch * TSEQ * 3;
    float* __restrict__ outb = out + (size_t)batch * TSEQ;
    float* __restrict__ lds = &xposeS[wv][0];

    // gate groups: cols 0-3 i, 4-7 f, 12-15 o (sigmoid); 8-11 g~ (tanh)
    const bool tanhLane = ((n >> 2) == 2);
    // Activation after WMMA:  act(v') = aMul * tanh(v') + bAdd, where the
    // sigmoid columns' 0.5 argument pre-scale is folded into W'/bias':
    //   tanh col:    scale 1.0, aMul = 1.0, bAdd = 0.0
    //   sigmoid col: scale 0.5, aMul = 0.5, bAdd = 0.5
    const float sScl = tanhLane ? 1.0f : 0.5f;
#if HAVE_HW_TANH
    const float aMul = tanhLane ? 1.0f : 0.5f;
    const float bAdd = tanhLane ? 0.0f : 0.5f;
#else
    // fallback: act = aMul * rcp(1 + exp2(eScl*v')) + bAdd  (v' pre-scaled)
    const float eScl = -2.0f * LOG2E;          // tanh(v') form for both
    const float aMul = tanhLane ? 2.0f : 1.0f;
    const float bAdd = tanhLane ? -1.0f : 0.0f;
#endif

    // ---- B operands (K x N = 4 x 16), pre-scaled by sScl per column:
    //      lane l, VGPR v holds B[K = 2*(l/16)+v][N = l%16]; B[k][n] = W[n][k].
    v2f Bih0, Bhh0, Bih1, Bhh1;
    Bih0.x = (k0 + 0 < 3) ? sScl * W_ih0[n * 3 + k0 + 0] : 0.0f;  // pad K=3
    Bih0.y = (k0 + 1 < 3) ? sScl * W_ih0[n * 3 + k0 + 1] : 0.0f;
    Bhh0.x = sScl * W_hh0[n * 4 + k0 + 0];
    Bhh0.y = sScl * W_hh0[n * 4 + k0 + 1];
    Bih1.x = sScl * W_ih1[n * 4 + k0 + 0];
    Bih1.y = sScl * W_ih1[n * 4 + k0 + 1];
    Bhh1.x = sScl * W_hh1[n * 4 + k0 + 0];
    Bhh1.y = sScl * W_hh1[n * 4 + k0 + 1];

    // pre-scaled bias splat as the C accumulator (per-column == per lane%16)
    const float bias0 = sScl * (b_ih0[n] + b_hh0[n]);
    const float bias1 = sScl * (b_ih1[n] + b_hh1[n]);
    v8f C0, C1;
    #pragma unroll
    for (int r = 0; r < 8; ++r) { C0[r] = bias0; C1[r] = bias1; }

    // head weights (tiny, uniform across lanes)
    float w1[16], b1v[4], w2[4];
    #pragma unroll
    for (int i = 0; i < 16; ++i) w1[i] = W1[i];
    #pragma unroll
    for (int i = 0; i < 4; ++i) { b1v[i] = b1[i]; w2[i] = W2[i]; }
    const float b2s = b2[0];

    // recurrent state in A layout: lane holds h[batch=n][k0], h[batch=n][k0+1]
    v2f hA0 = {0.0f, 0.0f}, c0 = {0.0f, 0.0f};
    v2f hA1 = {0.0f, 0.0f}, c1 = {0.0f, 0.0f};

    for (int t0 = 0; t0 < TSEQ; t0 += 4) {
        // keep the x cacheline stream warm (-> global_prefetch_b8)
        if ((t0 & 15) == 0 && t0 + 32 < TSEQ)
            __builtin_prefetch(xb + (t0 + 32) * 3, 0, 1);

        // 4 timesteps of x for this batch row: 12 floats, 16B aligned
        v4f q0 = *(const v4f*)(xb + t0 * 3 + 0);
        v4f q1 = *(const v4f*)(xb + t0 * 3 + 4);
        v4f q2 = *(const v4f*)(xb + t0 * 3 + 8);
        v4f yv;

        #pragma unroll
        for (int j = 0; j < 4; ++j) {
            // compile-time element indices; only the lane-half select is runtime
            auto getf = [&](int idx) -> float {
                return (idx < 4) ? q0[idx] : (idx < 8) ? q1[idx - 4] : q2[idx - 8];
            };
            const float e0 = getf(j * 3 + 0);
            const float e1 = getf(j * 3 + 1);
            const float e2 = getf(j * 3 + 2);
            float a0 = half ? e2 : e0;       // one v_cndmask
            float a1 = half ? 0.0f : e1;     // one v_cndmask (K=3 zero pad)
            v2f A0 = {a0, a1};

            // ================= layer 0 =================
            v8f d = __builtin_amdgcn_wmma_f32_16x16x4_f32(
                        false, A0,  false, Bih0, (short)0, C0, false, false);
            d = __builtin_amdgcn_wmma_f32_16x16x4_f32(
                        false, hA0, false, Bhh0, (short)0, d,  false, false);

            #pragma unroll
            for (int r = 0; r < 8; ++r) {
#if HAVE_HW_TANH
                d[r] = __fmaf_rn(aMul, __builtin_amdgcn_tanhf(d[r]), bAdd);
#else
                float e = __builtin_amdgcn_exp2f(d[r] * eScl);
                d[r] = __fmaf_rn(aMul, __builtin_amdgcn_rcpf(1.0f + e), bAdd);
#endif
            }
            #pragma unroll
            for (int r = 0; r < 8; ++r)
                lds[(r + 8 * half) * 16 + n] = d[r];
            __builtin_amdgcn_wave_barrier();
            v2f iP = *(const v2f*)&lds[n * 16 + 0  + k0];
            v2f fP = *(const v2f*)&lds[n * 16 + 4  + k0];
            v2f gP = *(const v2f*)&lds[n * 16 + 8  + k0];
            v2f oP = *(const v2f*)&lds[n * 16 + 12 + k0];
            __builtin_amdgcn_wave_barrier();

            c0.x = __fmaf_rn(fP.x, c0.x, iP.x * gP.x);
            c0.y = __fmaf_rn(fP.y, c0.y, iP.y * gP.y);
            hA0.x = oP.x * fast_tanh(c0.x);
            hA0.y = oP.y * fast_tanh(c0.y);

            // ================= layer 1 =================
            d = __builtin_amdgcn_wmma_f32_16x16x4_f32(
                        false, hA0, false, Bih1, (short)0, C1, false, false);
            d = __builtin_amdgcn_wmma_f32_16x16x4_f32(
                        false, hA1, false, Bhh1, (short)0, d,  false, false);

            #pragma unroll
            for (int r = 0; r < 8; ++r) {
#if HAVE_HW_TANH
                d[r] = __fmaf_rn(aMul, __builtin_amdgcn_tanhf(d[r]), bAdd);
#else
                float e = __builtin_amdgcn_exp2f(d[r] * eScl);
                d[r] = __fmaf_rn(aMul, __builtin_amdgcn_rcpf(1.0f + e), bAdd);
#endif
            }
            #pragma unroll
            for (int r = 0; r < 8; ++r)
                lds[(r + 8 * half) * 16 + n] = d[r];
            __builtin_amdgcn_wave_barrier();
            iP = *(const v2f*)&lds[n * 16 + 0  + k0];
            fP = *(const v2f*)&lds[n * 16 + 4  + k0];
            gP = *(const v2f*)&lds[n * 16 + 8  + k0];
            oP = *(const v2f*)&lds[n * 16 + 12 + k0];
            __builtin_amdgcn_wave_barrier();

            c1.x = __fmaf_rn(fP.x, c1.x, iP.x * gP.x);
            c1.y = __fmaf_rn(fP.y, c1.y, iP.y * gP.y);
            hA1.x = oP.x * fast_tanh(c1.x);
            hA1.y = oP.y * fast_tanh(c1.y);

            // ============ head: Linear(4,4) -> tanh -> Linear(4,1) =======
            float ox = __shfl_xor(hA1.x, 16, 32);
            float oy = __shfl_xor(hA1.y, 16, 32);
            float h0_ = half ? ox    : hA1.x;
            float h1_ = half ? oy    : hA1.y;
            float h2_ = half ? hA1.x : ox;
            float h3_ = half ? hA1.y : oy;

            float y = b2s;
            #pragma unroll
            for (int jj = 0; jj < 4; ++jj) {
                float z = b1v[jj];
                z = __fmaf_rn(w1[jj * 4 + 0], h0_, z);
                z = __fmaf_rn(w1[jj * 4 + 1], h1_, z);
                z = __fmaf_rn(w1[jj * 4 + 2], h2_, z);
                z = __fmaf_rn(w1[jj * 4 + 3], h3_, z);
                y = __fmaf_rn(w2[jj], fast_tanh(z), y);
            }
            yv[j] = y;
        }

        if (half == 0)
            *(v4f*)(outb + t0) = yv;   // 16B-aligned b128 store of 4 timesteps
    }
}

extern "C" void kernel_launch(void* const* d_in, const int* in_sizes, int n_in,
                              void* d_out, int out_size, void* d_ws, size_t ws_size,
                              hipStream_t stream) {
    (void)n_in; (void)out_size; (void)d_ws; (void)ws_size;
    const float* x     = (const float*)d_in[0];
    const float* W_ih0 = (const float*)d_in[1];
    const float* W_hh0 = (const float*)d_in[2];
    const float* b_ih0 = (const float*)d_in[3];
    const float* b_hh0 = (const float*)d_in[4];
    const float* W_ih1 = (const float*)d_in[5];
    const float* W_hh1 = (const float*)d_in[6];
    const float* b_ih1 = (const float*)d_in[7];
    const float* b_hh1 = (const float*)d_in[8];
    const float* W1    = (const float*)d_in[9];
    const float* b1    = (const float*)d_in[10];
    const float* W2    = (const float*)d_in[11];
    const float* b2    = (const float*)d_in[12];
    float* out = (float*)d_out;

    const int Btot = in_sizes[0] / (TSEQ * 3);   // 4096
    dim3 grid(Btot / 128);                        // 128 batches / 256-thread block
    dim3 block(256);                              // 8 waves x 16 batches
    lstm2_wmma_kernel<<<grid, block, 0, stream>>>(
        x, W_ih0, W_hh0, b_ih0, b_hh0,
        W_ih1, W_hh1, b_ih1, b_hh1,
        W1, b1, W2, b2, out);
}